// PositionalEmbeddings_17721035064077
// MI455X (gfx1250) — compile-verified
//
#include <hip/hip_runtime.h>

// B = 8, L = 4096, d_model = 1024 per the reference setup; L and d_model are
// derived from sizes on the host, BATCH is fixed by the reference.
#define BATCH 8
#define LOG2_10000 13.287712379549449

typedef int   v4i __attribute__((ext_vector_type(4)));
typedef float v4f __attribute__((ext_vector_type(4)));
typedef v4i __attribute__((address_space(1)))* gv4i_ptr;  // global int4*
typedef v4i __attribute__((address_space(3)))* lv4i_ptr;  // LDS int4*

#if defined(__has_builtin)
#if __has_builtin(__builtin_amdgcn_global_load_async_to_lds_b128)
#define HAVE_ASYNC_LDS 1
#endif
#endif

#ifndef HAVE_ASYNC_LDS
#define HAVE_ASYNC_LDS 0
#endif

__global__ __launch_bounds__(256) void pe_gather_add_kernel(
    const int* __restrict__ ids,     // [BATCH * L] token ids
    const float* __restrict__ emb,   // [vocab, dm]
    float* __restrict__ out,         // [BATCH, L, dm]
    int L, int dm,
    float step) {                    // step = -2*log2(10000)/dm (host-precomputed)
  extern __shared__ float smem[];    // BATCH * dm floats (32 KiB for dm=1024)

  const int l   = blockIdx.x;        // sequence position
  const int tid = threadIdx.x;
  const int stride = 4 * blockDim.x; // floats covered per iteration

  for (int d0 = 4 * tid; d0 < dm; d0 += stride) {
    // ---- issue the 8 data-dependent gathers (one b128 per lane per batch) ----
#if HAVE_ASYNC_LDS
#pragma unroll
    for (int b = 0; b < BATCH; ++b) {
      const int tok = ids[b * L + l];  // block-uniform -> scalar load
      const float* src = emb + (size_t)tok * (size_t)dm + d0;
      __builtin_amdgcn_global_load_async_to_lds_b128(
          (gv4i_ptr)src,                       // global source (AS1)
          (lv4i_ptr)&smem[b * dm + d0],        // LDS destination (AS3)
          /*offset=*/0, /*cpol=*/0);
    }
#else
    v4f v[BATCH];
#pragma unroll
    for (int b = 0; b < BATCH; ++b) {
      const int tok = ids[b * L + l];
      v[b] = *(const v4f*)(emb + (size_t)tok * (size_t)dm + d0);
    }
#endif

    // ---- compute the positional encoding while the loads are in flight ----
    // pe[l, 2k]   = sin(l * 2^(k*step)),  pe[l, 2k+1] = cos(l * 2^(k*step))
    // with step = -2*log2(10000)/dm  =>  2^(k*step) = 10000^(-2k/dm)
    const float pos = (float)l;
    const int   k0  = d0 >> 1;                 // pair index of elements d0, d0+1
    const float a0 = pos * exp2f((float)k0 * step);
    const float a1 = pos * exp2f((float)(k0 + 1) * step);
    float s0, c0, s1, c1;
    __sincosf(a0, &s0, &c0);
    __sincosf(a1, &s1, &c1);
    v4f pe;
    pe.x = s0; pe.y = c0; pe.z = s1; pe.w = c1;

#if HAVE_ASYNC_LDS
    // wait for all async global->LDS transfers from this wave
#if __has_builtin(__builtin_amdgcn_s_wait_asynccnt)
    __builtin_amdgcn_s_wait_asynccnt(0);
#else
    asm volatile("s_wait_asynccnt 0" ::: "memory");
#endif
    asm volatile("" ::: "memory");  // don't let LDS reads move above the wait
#endif

    // ---- add PE and stream out (non-temporal b128 stores) ----
#pragma unroll
    for (int b = 0; b < BATCH; ++b) {
#if HAVE_ASYNC_LDS
      const v4f t = *(const v4f*)&smem[b * dm + d0];  // own staged bytes
#else
      const v4f t = v[b];
#endif
      const v4f o = t + pe;
      v4f* dst = (v4f*)(out + ((size_t)b * (size_t)L + (size_t)l) * (size_t)dm + d0);
      __builtin_nontemporal_store(o, dst);  // write-once output: keep L2 for the gathers
    }
  }
}

extern "C" void kernel_launch(void* const* d_in, const int* in_sizes, int n_in,
                              void* d_out, int out_size, void* d_ws, size_t ws_size,
                              hipStream_t stream) {
  const int*   ids = (const int*)d_in[0];    // input_ids  [B, L] int32
  const float* emb = (const float*)d_in[1];  // emb_table  [vocab, dm] fp32
  float*       out = (float*)d_out;          // [B, L, dm] fp32
  (void)d_ws; (void)ws_size; (void)n_in;

  const int total_rows = in_sizes[0];        // B * L = 32768
  const int dm = out_size / total_rows;      // 1024
  const int L  = total_rows / BATCH;         // 4096
  const float step = (float)(-2.0 * LOG2_10000 / (double)dm);

  const dim3 grid(L);
  const dim3 block(256);                     // 256 threads * float4 = dm floats
  const size_t shmem = (size_t)BATCH * (size_t)dm * sizeof(float);  // 32 KiB

  pe_gather_add_kernel<<<grid, block, shmem, stream>>>(ids, emb, out, L, dm, step);
}